// SHOTWrapper_70239895158808
// MI455X (gfx1250) — compile-verified
//
#include <hip/hip_runtime.h>
#include <math.h>

// Problem sizes (fixed by the reference)
#define N_ROWS 16384
#define D_DIM  2048
#define H_DIM  512
#define C_CLS  65
#define C_PAD  80   // 5 WMMA col tiles of 16

typedef float v8f __attribute__((ext_vector_type(8)));
typedef float v2f __attribute__((ext_vector_type(2)));

// ---------------------------------------------------------------------------
// CDNA5 async global->LDS copy (ASYNCcnt-tracked). GVS addressing:
//   mem = SADDR(64b SGPR pair) + VADDR(32b unsigned byte offset)
// vdst VGPR carries the workgroup-relative LDS byte address.
// ---------------------------------------------------------------------------
__device__ __forceinline__ unsigned lds_addr_of(const void* p) {
    // flat shared pointer = {aperture_hi32, lds_offset32} -> truncate
    return (unsigned)(unsigned long long)p;
}
__device__ __forceinline__ void async_load_b128(unsigned lds_byte_addr,
                                                const void* base,
                                                unsigned byte_off) {
    asm volatile("global_load_async_to_lds_b128 %0, %1, %2"
                 :
                 : "v"(lds_byte_addr), "v"(byte_off), "s"(base)
                 : "memory");
}
__device__ __forceinline__ void wait_async_all() {
    asm volatile("s_wait_asynccnt 0x0" ::: "memory");
}

// ---------------------------------------------------------------------------
// Threefry-2x32 (20 rounds), key = jax.random.key(42) -> (k0,k1) = (0,42).
// Replicates jax.random.bernoulli(key, 0.5, (N,H)) bit layout.
// ---------------------------------------------------------------------------
__device__ __forceinline__ unsigned rotl32(unsigned x, int r) {
    return (x << r) | (x >> (32 - r));
}

__device__ __forceinline__ unsigned threefry2x32(unsigned x0, unsigned x1, int pick) {
    const unsigned k0 = 0u, k1 = 42u;
    const unsigned ks[3] = { k0, k1, 0x1BD11BDAu ^ k0 ^ k1 };
    const int R0[4] = {13, 15, 26, 6};
    const int R1[4] = {17, 29, 16, 24};
    x0 += ks[0]; x1 += ks[1];
    #pragma unroll
    for (int b = 0; b < 5; ++b) {
        #pragma unroll
        for (int j = 0; j < 4; ++j) {
            int r = (b & 1) ? R1[j] : R0[j];
            x0 += x1; x1 = rotl32(x1, r); x1 ^= x0;
        }
        x0 += ks[(b + 1) % 3];
        x1 += ks[(b + 2) % 3] + (unsigned)(b + 1);
    }
    return pick ? x1 : x0;
}

__device__ __forceinline__ float dropout_mask_scale(size_t idx) {
    const unsigned HALF = (unsigned)((size_t)N_ROWS * H_DIM / 2); // 4194304
    unsigned x0, x1; int pick;
    if ((unsigned)idx < HALF) { x0 = (unsigned)idx;        x1 = (unsigned)idx + HALF; pick = 0; }
    else                      { x0 = (unsigned)idx - HALF; x1 = (unsigned)idx;        pick = 1; }
    unsigned bits = threefry2x32(x0, x1, pick);
    float u = __uint_as_float((bits >> 9) | 0x3f800000u) - 1.0f;
    return (u < 0.5f) ? 2.0f : 0.0f;
}

// A-tile LDS stride: 36 floats = 144 B (16B-aligned rows for async B128;
// 36*m mod 64 is a permutation over m=0..15 -> conflict-free column reads).
#define AS_STRIDE 36
// B pair-interleaved stride: element (k,n) -> BsP[k>>1][n*2 + (k&1)].
// 160-float pair rows => the two wave halves hit disjoint bank sets.
#define BS_STRIDE 160

// ---------------------------------------------------------------------------
// Kernel 1: h = dropout(relu(feat @ w1 + b1))
// 256 thr = 8 waves; block tile 128 rows x 64 cols; K chunks of 32.
// A tile arrives via global_load_async_to_lds_b128 (overlapped with B fill).
// ---------------------------------------------------------------------------
__global__ __launch_bounds__(256)
void fc1_kernel(const float* __restrict__ feat, const float* __restrict__ w1,
                const float* __restrict__ b1, float* __restrict__ hbuf) {
    __shared__ float As[128][AS_STRIDE];
    __shared__ float BsP[16][BS_STRIDE];

    const int tid  = threadIdx.x;
    const int wave = tid >> 5;
    const int lane = tid & 31;
    const int m    = lane & 15;
    const int kh   = (lane >> 4) * 2;
    const int rowBase = (int)(blockIdx.x >> 3) * 128;  // 128 row blocks
    const int colBase = (int)(blockIdx.x & 7) * 64;    // 8 col blocks

    v8f acc[4] = {};

    for (int k0 = 0; k0 < D_DIM; k0 += 32) {
        // --- A tile: 128x32 floats = 1024 B128 transfers, 4 per thread (async)
        #pragma unroll
        for (int i = 0; i < 4; ++i) {
            int f = tid + i * 256;
            int r = f >> 3, c4 = (f & 7) << 2;
            async_load_b128(lds_addr_of(&As[r][c4]), feat,
                            (unsigned)(((size_t)(rowBase + r) * D_DIM + k0 + c4) * 4));
        }
        // --- B tile: 32x64, pair-interleaved into BsP (2 float4 per thread)
        #pragma unroll
        for (int i = 0; i < 2; ++i) {
            int f = tid + i * 256;
            int r = f >> 4, c4 = (f & 15) << 2;
            float4 v = *reinterpret_cast<const float4*>(
                &w1[(size_t)(k0 + r) * H_DIM + colBase + c4]);
            float* dst = &BsP[r >> 1][(r & 1)];
            dst[(c4 + 0) * 2] = v.x; dst[(c4 + 1) * 2] = v.y;
            dst[(c4 + 2) * 2] = v.z; dst[(c4 + 3) * 2] = v.w;
        }
        wait_async_all();
        __syncthreads();
        #pragma unroll
        for (int kk = 0; kk < 8; ++kk) {
            const int kb = kk * 4;
            const int pair = (kb + kh) >> 1;           // (kb+kh) is even
            v2f a, b;
            a.x = As[wave * 16 + m][kb + kh];
            a.y = As[wave * 16 + m][kb + kh + 1];
            #pragma unroll
            for (int t = 0; t < 4; ++t) {
                b = *reinterpret_cast<const v2f*>(&BsP[pair][(t * 16 + m) * 2]);
                acc[t] = __builtin_amdgcn_wmma_f32_16x16x4_f32(
                    false, a, false, b, (short)0, acc[t], false, false);
            }
        }
        __syncthreads();
    }

    const int half8 = (lane >> 4) * 8;
    #pragma unroll
    for (int t = 0; t < 4; ++t) {
        const int col = colBase + t * 16 + m;
        const float bias = b1[col];
        #pragma unroll
        for (int r = 0; r < 8; ++r) {
            const int row = rowBase + wave * 16 + half8 + r;
            float v = acc[t][r] + bias;
            v = v > 0.0f ? v : 0.0f;
            size_t idx = (size_t)row * H_DIM + col;
            hbuf[idx] = v * dropout_mask_scale(idx);
        }
    }
}

// ---------------------------------------------------------------------------
// Kernel 2: logits = h @ w2 + b2   (K=512, 65 cols padded to 80 = 5 tiles)
// ---------------------------------------------------------------------------
__global__ __launch_bounds__(256)
void fc2_kernel(const float* __restrict__ hbuf, const float* __restrict__ w2,
                const float* __restrict__ b2, float* __restrict__ logits) {
    __shared__ float As[128][AS_STRIDE];
    __shared__ float BsP[16][BS_STRIDE];

    const int tid  = threadIdx.x;
    const int wave = tid >> 5;
    const int lane = tid & 31;
    const int m    = lane & 15;
    const int kh   = (lane >> 4) * 2;
    const int rowBase = (int)blockIdx.x * 128;

    v8f acc[5] = {};

    for (int k0 = 0; k0 < H_DIM; k0 += 32) {
        #pragma unroll
        for (int i = 0; i < 4; ++i) {
            int f = tid + i * 256;
            int r = f >> 3, c4 = (f & 7) << 2;
            async_load_b128(lds_addr_of(&As[r][c4]), hbuf,
                            (unsigned)(((size_t)(rowBase + r) * H_DIM + k0 + c4) * 4));
        }
        // B tile: 32 rows x 80 cols (cols >= 65 zero), pair-interleaved
        #pragma unroll
        for (int i = 0; i < 10; ++i) {
            int f = tid + i * 256;            // f < 2560
            int r = f / 80, c = f % 80;
            float v = (c < C_CLS) ? w2[(size_t)(k0 + r) * C_CLS + c] : 0.0f;
            BsP[r >> 1][c * 2 + (r & 1)] = v;
        }
        wait_async_all();
        __syncthreads();
        #pragma unroll
        for (int kk = 0; kk < 8; ++kk) {
            const int kb = kk * 4;
            const int pair = (kb + kh) >> 1;
            v2f a, b;
            a.x = As[wave * 16 + m][kb + kh];
            a.y = As[wave * 16 + m][kb + kh + 1];
            #pragma unroll
            for (int t = 0; t < 5; ++t) {
                b = *reinterpret_cast<const v2f*>(&BsP[pair][(t * 16 + m) * 2]);
                acc[t] = __builtin_amdgcn_wmma_f32_16x16x4_f32(
                    false, a, false, b, (short)0, acc[t], false, false);
            }
        }
        __syncthreads();
    }

    const int half8 = (lane >> 4) * 8;
    #pragma unroll
    for (int t = 0; t < 5; ++t) {
        const int col = t * 16 + m;
        if (col < C_CLS) {
            const float bias = b2[col];
            #pragma unroll
            for (int r = 0; r < 8; ++r) {
                const int row = rowBase + wave * 16 + half8 + r;
                logits[(size_t)row * C_CLS + col] = acc[t][r] + bias;
            }
        }
    }
}

// ---------------------------------------------------------------------------
// Kernel 3: per-row softmax stats. One wave32 per row (65 cols = 32+32+1).
// ---------------------------------------------------------------------------
__global__ __launch_bounds__(256)
void softmax_kernel(const float* __restrict__ logits, float* __restrict__ rowmax,
                    float* __restrict__ lse, int* __restrict__ preds,
                    float* __restrict__ mp_acc, float* __restrict__ ent_acc) {
    __shared__ float smp[C_CLS];
    __shared__ float sent;
    const int tid  = threadIdx.x;
    const int warp = tid >> 5;
    const int lane = tid & 31;
    for (int i = tid; i < C_CLS; i += 256) smp[i] = 0.0f;
    if (tid == 0) sent = 0.0f;
    __syncthreads();

    const int row = (int)blockIdx.x * 8 + warp;   // grid = N/8
    const float NEG = -3.4e38f;
    float l0 = logits[(size_t)row * C_CLS + lane];
    float l1 = logits[(size_t)row * C_CLS + 32 + lane];
    float l2 = (lane == 0) ? logits[(size_t)row * C_CLS + 64] : NEG;

    float bv = l0; int bc = lane;
    if (l1 > bv) { bv = l1; bc = 32 + lane; }
    if (l2 > bv) { bv = l2; bc = 64; }
    float mx = bv;
    #pragma unroll
    for (int off = 16; off; off >>= 1) {
        float ov = __shfl_xor(bv, off, 32);
        int   oc = __shfl_xor(bc, off, 32);
        if (ov > bv || (ov == bv && oc < bc)) { bv = ov; bc = oc; }
        mx = fmaxf(mx, __shfl_xor(mx, off, 32));
    }
    float e0 = __expf(l0 - mx);
    float e1 = __expf(l1 - mx);
    float e2 = (lane == 0) ? __expf(l2 - mx) : 0.0f;
    float s = e0 + e1 + e2;
    #pragma unroll
    for (int off = 16; off; off >>= 1) s += __shfl_xor(s, off, 32);
    const float inv = 1.0f / s;

    float p0 = e0 * inv, p1 = e1 * inv, p2 = e2 * inv;
    float ent = p0 * logf(p0 + 1e-6f) + p1 * logf(p1 + 1e-6f);
    if (lane == 0) ent += p2 * logf(p2 + 1e-6f);
    #pragma unroll
    for (int off = 16; off; off >>= 1) ent += __shfl_xor(ent, off, 32);

    atomicAdd(&smp[lane], p0);
    atomicAdd(&smp[32 + lane], p1);
    if (lane == 0) atomicAdd(&smp[64], p2);
    if (lane == 0) {
        rowmax[row] = mx;
        lse[row]    = logf(s);
        preds[row]  = bc;
        atomicAdd(&sent, ent);
    }
    __syncthreads();
    if (tid < C_CLS) atomicAdd(&mp_acc[tid], smp[tid]);
    if (tid == 0)    atomicAdd(ent_acc, sent);
}

// ---------------------------------------------------------------------------
// Kernel 4: segment sums of feat rows into sums[C][D], counts[C].
// ---------------------------------------------------------------------------
__global__ __launch_bounds__(256)
void segsum_kernel(const float* __restrict__ feat, const int* __restrict__ preds,
                   float* __restrict__ sums, float* __restrict__ counts) {
    const int row = (int)blockIdx.x;
    const int cls = preds[row];
    for (int c = threadIdx.x; c < D_DIM; c += 256)
        atomicAdd(&sums[(size_t)cls * D_DIM + c], feat[(size_t)row * D_DIM + c]);
    if (threadIdx.x == 0) atomicAdd(&counts[cls], 1.0f);
}

// ---------------------------------------------------------------------------
// Kernel 5: centroids -> centT[D][C_PAD] (transposed, zero-padded, normalized)
// ---------------------------------------------------------------------------
__global__ __launch_bounds__(256)
void centroid_kernel(const float* __restrict__ sums, const float* __restrict__ counts,
                     float* __restrict__ centT) {
    __shared__ float red[256];
    const int c   = (int)blockIdx.x;  // grid = 65
    const int tid = threadIdx.x;
    const float cnt   = counts[c];
    const float denom = fmaxf(cnt, 1.0f);
    float ss = 0.0f;
    for (int d = tid; d < D_DIM; d += 256) {
        float mval = sums[(size_t)c * D_DIM + d] / denom;
        ss += mval * mval;
    }
    red[tid] = ss;
    __syncthreads();
    for (int off = 128; off; off >>= 1) {
        if (tid < off) red[tid] += red[tid + off];
        __syncthreads();
    }
    const float nrm = sqrtf(red[0]);
    const float scale = (cnt > 0.0f) ? 1.0f / (denom * fmaxf(nrm, 1e-12f)) : 0.0f;
    for (int d = tid; d < D_DIM; d += 256)
        centT[(size_t)d * C_PAD + c] = sums[(size_t)c * D_DIM + d] * scale;
}

// ---------------------------------------------------------------------------
// Kernel 6: sim = feat @ centT (argmax invariant to feat row norm),
// fused in-wave argmax over 65 cols -> pseudo_labels.
// ---------------------------------------------------------------------------
__global__ __launch_bounds__(256)
void sim_kernel(const float* __restrict__ feat, const float* __restrict__ centT,
                int* __restrict__ labels) {
    __shared__ float As[128][AS_STRIDE];
    __shared__ float BsP[16][BS_STRIDE];

    const int tid  = threadIdx.x;
    const int wave = tid >> 5;
    const int lane = tid & 31;
    const int m    = lane & 15;
    const int kh   = (lane >> 4) * 2;
    const int rowBase = (int)blockIdx.x * 128;

    v8f acc[5] = {};

    for (int k0 = 0; k0 < D_DIM; k0 += 32) {
        #pragma unroll
        for (int i = 0; i < 4; ++i) {
            int f = tid + i * 256;
            int r = f >> 3, c4 = (f & 7) << 2;
            async_load_b128(lds_addr_of(&As[r][c4]), feat,
                            (unsigned)(((size_t)(rowBase + r) * D_DIM + k0 + c4) * 4));
        }
        #pragma unroll
        for (int i = 0; i < 10; ++i) {
            int f = tid + i * 256;
            int r = f / 80, c = f % 80;
            BsP[r >> 1][c * 2 + (r & 1)] = centT[(size_t)(k0 + r) * C_PAD + c];
        }
        wait_async_all();
        __syncthreads();
        #pragma unroll
        for (int kk = 0; kk < 8; ++kk) {
            const int kb = kk * 4;
            const int pair = (kb + kh) >> 1;
            v2f a, b;
            a.x = As[wave * 16 + m][kb + kh];
            a.y = As[wave * 16 + m][kb + kh + 1];
            #pragma unroll
            for (int t = 0; t < 5; ++t) {
                b = *reinterpret_cast<const v2f*>(&BsP[pair][(t * 16 + m) * 2]);
                acc[t] = __builtin_amdgcn_wmma_f32_16x16x4_f32(
                    false, a, false, b, (short)0, acc[t], false, false);
            }
        }
        __syncthreads();
    }

    const int half8 = (lane >> 4) * 8;
    const float NEG = -3.4e38f;
    #pragma unroll
    for (int r = 0; r < 8; ++r) {
        float bv = NEG; int bc = 0;
        #pragma unroll
        for (int t = 0; t < 5; ++t) {
            const int col = t * 16 + m;
            float v = (col < C_CLS) ? acc[t][r] : NEG;
            if (v > bv) { bv = v; bc = col; }
        }
        #pragma unroll
        for (int off = 8; off; off >>= 1) {  // stays within 16-lane half
            float ov = __shfl_xor(bv, off, 32);
            int   oc = __shfl_xor(bc, off, 32);
            if (ov > bv || (ov == bv && oc < bc)) { bv = ov; bc = oc; }
        }
        if (m == 0)
            labels[rowBase + wave * 16 + half8 + r] = bc;
    }
}

// ---------------------------------------------------------------------------
// Kernel 7: CE partial sums: logp[n, label] = logits - rowmax - lse
// ---------------------------------------------------------------------------
__global__ __launch_bounds__(256)
void ce_kernel(const float* __restrict__ logits, const float* __restrict__ rowmax,
               const float* __restrict__ lse, const int* __restrict__ labels,
               float* __restrict__ ce_acc) {
    __shared__ float red[256];
    const int tid = threadIdx.x;
    const int i   = (int)blockIdx.x * 256 + tid;  // grid = N/256
    const int l   = labels[i];
    red[tid] = logits[(size_t)i * C_CLS + l] - rowmax[i] - lse[i];
    __syncthreads();
    for (int off = 128; off; off >>= 1) {
        if (tid < off) red[tid] += red[tid + off];
        __syncthreads();
    }
    if (tid == 0) atomicAdd(ce_acc, red[0]);
}

// ---------------------------------------------------------------------------
// Kernel 8: combine scalars -> loss
// ---------------------------------------------------------------------------
__global__ void finalize_kernel(const float* __restrict__ mp_acc,
                                const float* __restrict__ ent_acc,
                                const float* __restrict__ ce_acc,
                                float* __restrict__ out) {
    if (threadIdx.x == 0 && blockIdx.x == 0) {
        const float invN = 1.0f / (float)N_ROWS;
        float entropy = -(*ent_acc) * invN;
        float div = 0.0f;
        for (int c = 0; c < C_CLS; ++c) {
            float mp = mp_acc[c] * invN;
            div += mp * logf(mp + 1e-6f);
        }
        float ce = -(*ce_acc) * invN;
        out[0] = entropy + div + 0.3f * ce;
    }
}

// ---------------------------------------------------------------------------
extern "C" void kernel_launch(void* const* d_in, const int* in_sizes, int n_in,
                              void* d_out, int out_size, void* d_ws, size_t ws_size,
                              hipStream_t stream) {
    (void)in_sizes; (void)n_in; (void)out_size; (void)ws_size;
    const float* feat = (const float*)d_in[0];
    const float* w1   = (const float*)d_in[1];
    const float* b1   = (const float*)d_in[2];
    const float* w2   = (const float*)d_in[3];
    const float* b2   = (const float*)d_in[4];
    float* out = (float*)d_out;

    char* p = (char*)d_ws;
    auto take = [&](size_t bytes) -> void* {
        void* r = (void*)p;
        p += (bytes + 255) & ~(size_t)255;
        return r;
    };
    float* hbuf    = (float*)take((size_t)N_ROWS * H_DIM * 4);   // 33.5 MB
    float* logits  = (float*)take((size_t)N_ROWS * C_CLS * 4);   // 4.3 MB
    float* rowmax  = (float*)take((size_t)N_ROWS * 4);
    float* lse     = (float*)take((size_t)N_ROWS * 4);
    int*   preds   = (int*)  take((size_t)N_ROWS * 4);
    int*   labels  = (int*)  take((size_t)N_ROWS * 4);
    float* sums    = (float*)take((size_t)C_CLS * D_DIM * 4);    // 532 KB
    float* counts  = (float*)take((size_t)C_CLS * 4);
    float* centT   = (float*)take((size_t)D_DIM * C_PAD * 4);    // 655 KB
    float* mp_acc  = (float*)take((size_t)C_CLS * 4);
    float* ent_acc = (float*)take(4);
    float* ce_acc  = (float*)take(4);

    // Zero accumulators every call (graph-capture legal).
    hipMemsetAsync(sums,    0, (size_t)C_CLS * D_DIM * 4, stream);
    hipMemsetAsync(counts,  0, (size_t)C_CLS * 4, stream);
    hipMemsetAsync(centT,   0, (size_t)D_DIM * C_PAD * 4, stream);
    hipMemsetAsync(mp_acc,  0, (size_t)C_CLS * 4, stream);
    hipMemsetAsync(ent_acc, 0, 4, stream);
    hipMemsetAsync(ce_acc,  0, 4, stream);

    fc1_kernel     <<<dim3(1024),        256, 0, stream>>>(feat, w1, b1, hbuf);
    fc2_kernel     <<<dim3(N_ROWS/128),  256, 0, stream>>>(hbuf, w2, b2, logits);
    softmax_kernel <<<dim3(N_ROWS/8),    256, 0, stream>>>(logits, rowmax, lse, preds, mp_acc, ent_acc);
    segsum_kernel  <<<dim3(N_ROWS),      256, 0, stream>>>(feat, preds, sums, counts);
    centroid_kernel<<<dim3(C_CLS),       256, 0, stream>>>(sums, counts, centT);
    sim_kernel     <<<dim3(N_ROWS/128),  256, 0, stream>>>(feat, centT, labels);
    ce_kernel      <<<dim3(N_ROWS/256),  256, 0, stream>>>(logits, rowmax, lse, labels, ce_acc);
    finalize_kernel<<<dim3(1),           64,  0, stream>>>(mp_acc, ent_acc, ce_acc, out);
}